// NullModel_77893526880666
// MI455X (gfx1250) — compile-verified
//
#include <hip/hip_runtime.h>

// DVF -> DDF scaling-and-squaring (MONAI DVF2DDF, 7 steps) + bilinear/nearest warps.
// MI455X strategy:
//  * field re-packed as interleaved float4/voxel -> every trilinear corner is one b128.
//  * per squaring step, each 256-thread block stages a haloed 24x24x16 sub-box of the
//    field into LDS with ONE Tensor Data Mover descriptor (tensor_load_to_lds,
//    TENSORcnt-tracked), then gathers corners with ds_load_b128; rare out-of-halo
//    lanes fall back to global gathers (exact border-clamp semantics preserved).
//  * global_prefetch_b8 on the streaming init pass, TH_NT on the dvf passthrough.
// Workspace: needs 2 * NVOX * 16 B = 64 MB in d_ws (ping-pong field buffers).

namespace {

constexpr int Dz = 128, Hy = 128, Wx = 128;
constexpr int NVOX   = Dz * Hy * Wx;     // 2,097,152
constexpr int NSTEPS = 7;

// d_out flat layout (reference return order): ddf[3*N], image[N], label[N], dvf[3*N]
constexpr int OUT_DDF = 0;
constexpr int OUT_IMG = 3 * NVOX;
constexpr int OUT_LBL = 4 * NVOX;
constexpr int OUT_DVF = 5 * NVOX;

// LDS-staging tile geometry for the squaring step.
constexpr int TX = 16, TY = 16, TZ = 8;  // voxels computed per block (2048, 8/thread)
constexpr int RAD = 4;                   // halo radius (covers |disp| < ~4 voxels)
constexpr int EX = TX + 2 * RAD;         // 24
constexpr int EY = TY + 2 * RAD;         // 24
constexpr int EZ = TZ + 2 * RAD;         // 16
constexpr int BOX = EX * EY * EZ;        // 9216 float4 = 147,456 B LDS

typedef __attribute__((ext_vector_type(4))) float        f4_t;
typedef __attribute__((ext_vector_type(4))) unsigned int u32x4;
typedef __attribute__((ext_vector_type(8))) int          i32x8;
typedef __attribute__((ext_vector_type(4))) int          i32x4;

__device__ __forceinline__ int vidx(int z, int y, int x) {
  return (z * Hy + y) * Wx + x;
}

__device__ __forceinline__ float4 f4lerp(const float4 a, const float4 b, const float t) {
  float4 r;
  r.x = fmaf(t, b.x - a.x, a.x);
  r.y = fmaf(t, b.y - a.y, a.y);
  r.z = fmaf(t, b.z - a.z, a.z);
  r.w = 0.0f;
  return r;
}

} // namespace

#if __has_builtin(__builtin_amdgcn_tensor_load_to_lds) && \
    __has_builtin(__builtin_amdgcn_s_wait_tensorcnt)
#define HAVE_TDM 1
#else
#define HAVE_TDM 0
#endif

// ---------------------------------------------------------------------------
// 1) init: ddf = dvf / 2^7, planar [3,D,H,W] -> interleaved float4 per voxel.
// ---------------------------------------------------------------------------
__global__ __launch_bounds__(256) void vf_init(const float* __restrict__ dvf,
                                               float4* __restrict__ fld) {
  const int stride = gridDim.x * blockDim.x;
  for (int i = blockIdx.x * blockDim.x + threadIdx.x; i < NVOX; i += stride) {
    __builtin_prefetch(dvf + i + stride, 0, 0);
    __builtin_prefetch(dvf + NVOX + i + stride, 0, 0);
    __builtin_prefetch(dvf + 2 * NVOX + i + stride, 0, 0);
    float4 v;
    v.x = dvf[i]            * (1.0f / 128.0f);  // channel 0 = z-displacement
    v.y = dvf[NVOX + i]     * (1.0f / 128.0f);  // channel 1 = y-displacement
    v.z = dvf[2 * NVOX + i] * (1.0f / 128.0f);  // channel 2 = x-displacement
    v.w = 0.0f;
    fld[i] = v;
  }
}

// ---------------------------------------------------------------------------
// 2) one squaring step with TDM-staged LDS halo box:
//    out = in + warp(in, in, bilinear)
// ---------------------------------------------------------------------------
__global__ __launch_bounds__(256) void vf_square_step_lds(const float4* __restrict__ in,
                                                          float4* __restrict__ out) {
  extern __shared__ float4 box[];   // [EZ][EY][EX]

  const int tsx = blockIdx.x * TX;
  const int tsy = blockIdx.y * TY;
  const int tsz = blockIdx.z * TZ;
  // Halo-box origin, shifted so the box is always fully interior (no OOB).
  const int ox = min(max(tsx - RAD, 0), Wx - EX);
  const int oy = min(max(tsy - RAD, 0), Hy - EY);
  const int oz = min(max(tsz - RAD, 0), Dz - EZ);

#if HAVE_TDM
  if (threadIdx.x < 32) {   // one wave issues the DMA; EXEC is ignored by TDM
    const unsigned long long gaddr =
        (unsigned long long)(uintptr_t)(const void*)(in + vidx(oz, oy, ox));
    const unsigned ldsAddr = (unsigned)(uintptr_t)(void*)box;  // low 32 = LDS offset

    // data_size = 8B  =>  x-dimension counted in 2-unit halves of a float4.
    const unsigned td0 = (unsigned)((Wx - ox) * 2);  // valid units from tile start, dim0
    const unsigned td1 = (unsigned)(Hy - oy);
    const unsigned td2 = (unsigned)(Dz - oz);
    const unsigned st0 = (unsigned)(Wx * 2);             // line stride, units
    const unsigned st1 = (unsigned)(Wx * Hy * 2);        // slice stride, units
    const unsigned st2 = (unsigned)(Wx * Hy * Dz * 2);   // volume stride, units

    u32x4 g0;
    g0[0] = 1u;                                          // count=1, load, no gather
    g0[1] = ldsAddr;                                     // lds_addr [63:32]
    g0[2] = (unsigned)(gaddr & 0xFFFFFFFFull);           // global_addr [95:64]
    g0[3] = (unsigned)((gaddr >> 32) & 0x01FFFFFFull)    // global_addr [120:96]
            | 0x80000000u;                               // type=2 ("image") [127:126]

    i32x8 g1;
    g1[0] = (int)(3u << 16);                             // wg_mask=0, data_size=3 (8B)
    g1[1] = (int)((td0 & 0xFFFFu) << 16);                // tensor_dim0[15:0] @ [63:48]
    g1[2] = (int)((td0 >> 16) | ((td1 & 0xFFFFu) << 16));// td0 hi | tensor_dim1 lo
    g1[3] = (int)((td1 >> 16) | ((unsigned)(EX * 2) << 16)); // td1 hi | tile_dim0
    g1[4] = (int)((unsigned)EY | ((unsigned)EZ << 16));  // tile_dim1 | tile_dim2
    g1[5] = (int)st0;                                    // tensor_dim0_stride[31:0]
    g1[6] = (int)((st1 & 0xFFFFu) << 16);                // st0 hi16=0 | st1[15:0]
    g1[7] = (int)(st1 >> 16);                            // st1[47:16]

    i32x4 g2;
    g2[0] = (int)td2;                                    // tensor_dim2
    g2[1] = 1;                                           // tensor_dim3 (unused)
    g2[2] = (int)st2;                                    // tensor_dim2_stride[31:0]
    g2[3] = 0;                                           // st2 hi=0, tile_dim3=0

    i32x4 g3;
    g3[0] = 0; g3[1] = 0; g3[2] = 0; g3[3] = 0;          // dims 4 unused

#if defined(__clang_major__) && (__clang_major__ >= 23)
    i32x8 g4;
    g4[0] = 0; g4[1] = 0; g4[2] = 0; g4[3] = 0;
    g4[4] = 0; g4[5] = 0; g4[6] = 0; g4[7] = 0;
    __builtin_amdgcn_tensor_load_to_lds(g0, g1, g2, g3, g4, 0);
#else
    __builtin_amdgcn_tensor_load_to_lds(g0, g1, g2, g3, 0);
#endif
    __builtin_amdgcn_s_wait_tensorcnt(0);
  }
  __syncthreads();
#else
  // Cooperative staging fallback (no TDM builtin on this toolchain).
  for (int t = threadIdx.x; t < BOX; t += 256) {
    const int bx = t % EX;
    const int r  = t / EX;
    const int by = r % EY;
    const int bz = r / EY;
    box[t] = in[vidx(oz + bz, oy + by, ox + bx)];
  }
  __syncthreads();
#endif

  const int tx = threadIdx.x & (TX - 1);
  const int ty = (threadIdx.x >> 4) & (TY - 1);
  const int x = tsx + tx;
  const int y = tsy + ty;

  for (int s = 0; s < TZ; ++s) {
    const int z = tsz + s;
    const int i = vidx(z, y, x);

    // own displacement straight from the staged box
    const float4 u = box[((z - oz) * EY + (y - oy)) * EX + (x - ox)];

    const float fz = fminf(fmaxf((float)z + u.x, 0.0f), (float)(Dz - 1));
    const float fy = fminf(fmaxf((float)y + u.y, 0.0f), (float)(Hy - 1));
    const float fx = fminf(fmaxf((float)x + u.z, 0.0f), (float)(Wx - 1));

    const float zf = floorf(fz), yf = floorf(fy), xf = floorf(fx);
    const int z0 = (int)zf, y0 = (int)yf, x0 = (int)xf;
    const int z1 = min(z0 + 1, Dz - 1);
    const int y1 = min(y0 + 1, Hy - 1);
    const int x1 = min(x0 + 1, Wx - 1);
    const float wz = fz - zf, wy = fy - yf, wx = fx - xf;

    float4 c000, c001, c010, c011, c100, c101, c110, c111;
    const bool inbox = (x0 >= ox) & (x1 < ox + EX) &
                       (y0 >= oy) & (y1 < oy + EY) &
                       (z0 >= oz) & (z1 < oz + EZ);
    if (inbox) {
      const int bz0 = z0 - oz, bz1 = z1 - oz;
      const int by0 = y0 - oy, by1 = y1 - oy;
      const int bx0 = x0 - ox, bx1 = x1 - ox;
      c000 = box[(bz0 * EY + by0) * EX + bx0];
      c001 = box[(bz0 * EY + by0) * EX + bx1];
      c010 = box[(bz0 * EY + by1) * EX + bx0];
      c011 = box[(bz0 * EY + by1) * EX + bx1];
      c100 = box[(bz1 * EY + by0) * EX + bx0];
      c101 = box[(bz1 * EY + by0) * EX + bx1];
      c110 = box[(bz1 * EY + by1) * EX + bx0];
      c111 = box[(bz1 * EY + by1) * EX + bx1];
    } else {
      c000 = in[vidx(z0, y0, x0)];
      c001 = in[vidx(z0, y0, x1)];
      c010 = in[vidx(z0, y1, x0)];
      c011 = in[vidx(z0, y1, x1)];
      c100 = in[vidx(z1, y0, x0)];
      c101 = in[vidx(z1, y0, x1)];
      c110 = in[vidx(z1, y1, x0)];
      c111 = in[vidx(z1, y1, x1)];
    }

    const float4 a0 = f4lerp(c000, c001, wx);
    const float4 a1 = f4lerp(c010, c011, wx);
    const float4 a2 = f4lerp(c100, c101, wx);
    const float4 a3 = f4lerp(c110, c111, wx);
    const float4 b0 = f4lerp(a0, a1, wy);
    const float4 b1 = f4lerp(a2, a3, wy);
    const float4 r  = f4lerp(b0, b1, wz);

    float4 o;
    o.x = u.x + r.x;
    o.y = u.y + r.y;
    o.z = u.z + r.z;
    o.w = 0.0f;
    out[i] = o;
  }
}

// ---------------------------------------------------------------------------
// 3) finalize: emit planar ddf, bilinear-warped image, nearest-warped label.
// ---------------------------------------------------------------------------
__global__ __launch_bounds__(256) void vf_finalize(const float4* __restrict__ fld,
                                                   const float* __restrict__ img,
                                                   const float* __restrict__ lbl,
                                                   float* __restrict__ out) {
  const int i = blockIdx.x * 256 + threadIdx.x;
  const int x = i & (Wx - 1);
  const int y = (i >> 7) & (Hy - 1);
  const int z = i >> 14;

  const float4 u = fld[i];
  out[OUT_DDF + i]            = u.x;
  out[OUT_DDF + NVOX + i]     = u.y;
  out[OUT_DDF + 2 * NVOX + i] = u.z;

  const float fz = fminf(fmaxf((float)z + u.x, 0.0f), (float)(Dz - 1));
  const float fy = fminf(fmaxf((float)y + u.y, 0.0f), (float)(Hy - 1));
  const float fx = fminf(fmaxf((float)x + u.z, 0.0f), (float)(Wx - 1));

  const float zf = floorf(fz), yf = floorf(fy), xf = floorf(fx);
  const int z0 = (int)zf, y0 = (int)yf, x0 = (int)xf;
  const int z1 = min(z0 + 1, Dz - 1);
  const int y1 = min(y0 + 1, Hy - 1);
  const int x1 = min(x0 + 1, Wx - 1);
  const float wz = fz - zf, wy = fy - yf, wx = fx - xf;

  const float v000 = img[vidx(z0, y0, x0)];
  const float v001 = img[vidx(z0, y0, x1)];
  const float v010 = img[vidx(z0, y1, x0)];
  const float v011 = img[vidx(z0, y1, x1)];
  const float v100 = img[vidx(z1, y0, x0)];
  const float v101 = img[vidx(z1, y0, x1)];
  const float v110 = img[vidx(z1, y1, x0)];
  const float v111 = img[vidx(z1, y1, x1)];

  const float a0 = fmaf(wx, v001 - v000, v000);
  const float a1 = fmaf(wx, v011 - v010, v010);
  const float a2 = fmaf(wx, v101 - v100, v100);
  const float a3 = fmaf(wx, v111 - v110, v110);
  const float b0 = fmaf(wy, a1 - a0, a0);
  const float b1 = fmaf(wy, a3 - a2, a2);
  out[OUT_IMG + i] = fmaf(wz, b1 - b0, b0);

  const int zi = (int)rintf(fz);
  const int yi = (int)rintf(fy);
  const int xi = (int)rintf(fx);
  out[OUT_LBL + i] = lbl[vidx(zi, yi, xi)];
}

// ---------------------------------------------------------------------------
// 4) dvf passthrough, b128 non-temporal streaming copy.
// ---------------------------------------------------------------------------
__global__ __launch_bounds__(256) void vf_copy_dvf(const f4_t* __restrict__ src,
                                                   f4_t* __restrict__ dst) {
  const int i = blockIdx.x * 256 + threadIdx.x;
  if (i < (3 * NVOX) / 4) {
    const f4_t v = __builtin_nontemporal_load(&src[i]);
    __builtin_nontemporal_store(v, &dst[i]);
  }
}

extern "C" void kernel_launch(void* const* d_in, const int* in_sizes, int n_in,
                              void* d_out, int out_size, void* d_ws, size_t ws_size,
                              hipStream_t stream) {
  (void)in_sizes; (void)n_in; (void)out_size; (void)ws_size;

  const float* dvf = (const float*)d_in[0];   // [1,3,128,128,128]
  const float* img = (const float*)d_in[1];   // moving_image
  // d_in[2] (fixed_image) is unused by the reference outputs.
  const float* lbl = (const float*)d_in[3];   // moving_label
  float* out = (float*)d_out;

  float4* bufA = (float4*)d_ws;               // 32 MB
  float4* bufB = bufA + NVOX;                 // 32 MB

  vf_init<<<2048, 256, 0, stream>>>(dvf, bufA);

  const dim3 tiles(Wx / TX, Hy / TY, Dz / TZ);   // 8 x 8 x 16 blocks
  const size_t ldsBytes = (size_t)BOX * sizeof(float4);

  const float4* src = bufA;
  float4* dst = bufB;
  for (int s = 0; s < NSTEPS; ++s) {
    vf_square_step_lds<<<tiles, 256, ldsBytes, stream>>>(src, dst);
    const float4* t = dst;
    dst = (float4*)src;
    src = t;
  }
  // after 7 steps the final field is pointed to by `src`

  vf_finalize<<<NVOX / 256, 256, 0, stream>>>(src, img, lbl, out);
  vf_copy_dvf<<<(3 * NVOX / 4) / 256, 256, 0, stream>>>((const f4_t*)dvf,
                                                        (f4_t*)(out + OUT_DVF));
}